// Attention_block_78915729097159
// MI455X (gfx1250) — compile-verified
//
#include <hip/hip_runtime.h>

typedef __attribute__((ext_vector_type(2))) float v2f;
typedef __attribute__((ext_vector_type(8))) float v8f;

#define BATCH 4
#define CH    64
#define WD    64
#define HW    4096   // 64*64

static __device__ __forceinline__ v8f wmma_f32_k4(v2f a, v2f b, v8f c) {
  // D = A(16x4 f32) * B(4x16 f32) + C(16x16 f32)
  return __builtin_amdgcn_wmma_f32_16x16x4_f32(
      /*neg_a=*/false, a, /*neg_b=*/false, b,
      /*c_mod=*/(short)0, c, /*reuse_a=*/false, /*reuse_b=*/false);
}

// Async copy: global -> LDS, tracked by ASYNCcnt.
// VDST operand = LDS byte address (flat->LDS mapping truncates to addr[31:0]).
static __device__ __forceinline__ void async_g2l_b128(const void* lds_ptr,
                                                      const void* gptr) {
  unsigned int loff = (unsigned int)(unsigned long long)lds_ptr;
  asm volatile("global_load_async_to_lds_b128 %0, %1, off"
               :: "v"(loff), "v"(gptr) : "memory");
}
static __device__ __forceinline__ void async_g2l_b32(const void* lds_ptr,
                                                     const void* gptr) {
  unsigned int loff = (unsigned int)(unsigned long long)lds_ptr;
  asm volatile("global_load_async_to_lds_b32 %0, %1, off"
               :: "v"(loff), "v"(gptr) : "memory");
}
static __device__ __forceinline__ void wait_async0() {
  asm volatile("s_wait_asynccnt 0x0" ::: "memory");
}

// ---------------------------------------------------------------------------
// Kernel 1: fused score-GEMM + row max/argmax.
// T[i,j] = sum_c A[c,i] * V[c,j];  S[j] = max_i T, IDX[j] = argmax_i T.
// 8 waves/block each own a 16-column j-tile; the 16-row A-tile (4 KB) is
// staged once per block into double-buffered LDS via async Global->LDS DMA
// (8x cut in L2 A-traffic). Per tile: ALL 16 DS operand loads are issued
// first into a register array (one bulk DScnt wait), then the 16 chained
// WMMAs run as an uninterrupted XDL burst. T never leaves VGPRs.
// ---------------------------------------------------------------------------
__global__ __launch_bounds__(256) void hard_match_kernel(
    const float* __restrict__ Afeat, const float* __restrict__ Vfeat,
    float* __restrict__ S, int* __restrict__ IDX)
{
  __shared__ float Asm[2][CH][16];               // 2 x 4 KB ping-pong

  const int tid   = threadIdx.x;
  const int lane  = tid & 31;
  const int wave  = tid >> 5;
  const int b     = blockIdx.x >> 5;             // 32 j-groups per batch
  const int jbase = (((blockIdx.x & 31) << 3) + wave) << 4;
  const int half  = lane >> 4;                   // lane group (K split)
  const int lh    = lane & 15;

  const float* Ab = Afeat + b * (CH * HW);
  const float* Vb = Vfeat + b * (CH * HW);

  // Staging role: each of 256 threads moves one 16B chunk per tile.
  const int sc = tid >> 2;                       // channel 0..63
  const int si = (tid & 3) << 2;                 // i-offset 0,4,8,12

  // Preload B operand: 16 k-chunks of the 16-column V slab (reg-resident).
  v2f Breg[16];
#pragma unroll
  for (int kc = 0; kc < 16; ++kc) {
    const int c = kc * 4 + half * 2;
    Breg[kc].x = Vb[c * HW + jbase + lh];
    Breg[kc].y = Vb[(c + 1) * HW + jbase + lh];
  }

  // Prologue: stage tile 0.
  async_g2l_b128(&Asm[0][sc][si], Ab + sc * HW + si);
  wait_async0();
  __syncthreads();

  float best  = -3.402823466e+38f;
  int   bestI = 0;

  for (int it = 0; it < 256; ++it) {
    const int ibase = it << 4;
    const int cur   = it & 1;
    // Prefetch next A-tile into the other buffer (async, overlaps WMMA).
    if (it + 1 < 256)
      async_g2l_b128(&Asm[cur ^ 1][sc][si], Ab + sc * HW + ((it + 1) << 4) + si);

    // Bulk-load the A-tile operands from LDS (16 ds_load_2addr in flight).
    v2f a[16];
#pragma unroll
    for (int kc = 0; kc < 16; ++kc) {
      const int c = kc * 4 + half * 2;
      a[kc].x = Asm[cur][c][lh];
      a[kc].y = Asm[cur][c + 1][lh];
    }

    // Uninterrupted chained WMMA burst.
    v8f acc = {0, 0, 0, 0, 0, 0, 0, 0};
#pragma unroll
    for (int kc = 0; kc < 16; ++kc)
      acc = wmma_f32_k4(a[kc], Breg[kc], acc);

    // Per-lane max over its 8 rows (rows M = r + 8*half; '>' keeps first).
    float lb = acc[0];
    int   li = ibase + 8 * half;
#pragma unroll
    for (int r = 1; r < 8; ++r) {
      const float v = acc[r];
      if (v > lb) { lb = v; li = ibase + r + 8 * half; }
    }
    // Merge with the other lane-half (same column, other rows); ties -> lower i.
    const float pv = __shfl_xor(lb, 16, 32);
    const int   pi = __shfl_xor(li, 16, 32);
    if (pv > lb || (pv == lb && pi < li)) { lb = pv; li = pi; }
    if (lb > best) { best = lb; bestI = li; }

    // My async writes done + everyone finished reading cur buffer.
    wait_async0();
    __syncthreads();
  }

  if (half == 0) {
    S[b * HW + jbase + lh]   = best;
    IDX[b * HW + jbase + lh] = bestI;
  }
}

// ---------------------------------------------------------------------------
// Kernel 2: conv1 (1x1, 2C->C) over concat([V, gather(A, idx)]), scaled by S.
// Implicit GEMM: M = out-channels (4 tiles of 16), N = 16 pixels, K = 128.
// ---------------------------------------------------------------------------
__global__ __launch_bounds__(256) void conv1_fuse_kernel(
    const float* __restrict__ Vfeat, const float* __restrict__ Afeat,
    const int* __restrict__ IDX, const float* __restrict__ S,
    const float* __restrict__ w1, const float* __restrict__ b1,
    float* __restrict__ Out)
{
  const int lane  = threadIdx.x & 31;
  const int wave  = threadIdx.x >> 5;
  const int wt    = blockIdx.x * 8 + wave;
  const int b     = wt >> 8;
  const int jbase = (wt & 255) << 4;
  const int half  = lane >> 4;
  const int lh    = lane & 15;

  const float* Vb = Vfeat + b * (CH * HW);
  const float* Ab = Afeat + b * (CH * HW);
  const int   myIdx = IDX[b * HW + jbase + lh];
  const float myS   = S[b * HW + jbase + lh];

  v8f acc[4] = {{0,0,0,0,0,0,0,0},{0,0,0,0,0,0,0,0},
                {0,0,0,0,0,0,0,0},{0,0,0,0,0,0,0,0}};

#pragma unroll
  for (int kc = 0; kc < 32; ++kc) {
    const int k0 = kc * 4 + half * 2;  // boundary at 64 falls on a chunk edge
    v2f bv;  // B operand = per-pixel feature vector (V channels then gathered A)
    bv.x = (k0     < CH) ? Vb[k0 * HW + jbase + lh]
                         : Ab[(k0 - CH) * HW + myIdx];
    bv.y = (k0 + 1 < CH) ? Vb[(k0 + 1) * HW + jbase + lh]
                         : Ab[(k0 + 1 - CH) * HW + myIdx];
#pragma unroll
    for (int ot = 0; ot < 4; ++ot) {
      const int o = ot * 16 + lh;      // A operand = weights, M = out-channel
      v2f av;
      av.x = w1[o * (2 * CH) + k0];
      av.y = w1[o * (2 * CH) + k0 + 1];
      acc[ot] = wmma_f32_k4(av, bv, acc[ot]);
    }
  }

#pragma unroll
  for (int ot = 0; ot < 4; ++ot) {
#pragma unroll
    for (int r = 0; r < 8; ++r) {
      const int o = ot * 16 + r + 8 * half;
      Out[(b * CH + o) * HW + jbase + lh] = (acc[ot][r] + b1[o]) * myS;
    }
  }
}

// ---------------------------------------------------------------------------
// Kernel 3: conv2 (3x3, 3C->C, SAME) over [C_v, C_k, V] + eval BN + ReLU.
// Implicit GEMM: M = out-channels, N = 16 pixels of one image row, K = 9*192.
// Per-tap weight slice (192x64 = 48 KB) staged into LDS once per block
// (async DMA), transposed to [cin][cout] with a 72-float padded stride so
// the two lane-halves read disjoint LDS banks. Operand loads are hoisted
// ahead of each 4-WMMA group.
// ---------------------------------------------------------------------------
#define WSTRIDE 72

static __device__ __forceinline__ float fused_ld(
    const float* __restrict__ Cv, const float* __restrict__ Ck,
    const float* __restrict__ Vb, int c, int p)
{
  if (c < CH)      return Cv[c * HW + p];
  if (c < 2 * CH)  return Ck[(c - CH) * HW + p];
  return Vb[(c - 2 * CH) * HW + p];
}

__global__ __launch_bounds__(256) void conv2_bn_relu_kernel(
    const float* __restrict__ Cv, const float* __restrict__ Ck,
    const float* __restrict__ Vfeat, const float* __restrict__ w2,
    const float* __restrict__ b2, const float* __restrict__ gamma,
    const float* __restrict__ beta, const float* __restrict__ mean,
    const float* __restrict__ var, float* __restrict__ Out)
{
  __shared__ float Wlds[3 * CH * WSTRIDE];   // 192 x 72 floats = 54 KB

  const int tid  = threadIdx.x;
  const int lane = tid & 31;
  const int wave = tid >> 5;
  const int wt   = blockIdx.x * 8 + wave;
  const int b    = wt >> 8;
  const int pt   = wt & 255;        // pixel tile (16 consecutive, same row x)
  const int x    = pt >> 2;
  const int y0   = (pt & 3) << 4;
  const int half = lane >> 4;
  const int lh   = lane & 15;
  const int y    = y0 + lh;

  const float* CvB = Cv    + b * (CH * HW);
  const float* CkB = Ck    + b * (CH * HW);
  const float* VB  = Vfeat + b * (CH * HW);

  v8f acc[4] = {{0,0,0,0,0,0,0,0},{0,0,0,0,0,0,0,0},
                {0,0,0,0,0,0,0,0},{0,0,0,0,0,0,0,0}};

  for (int tap = 0; tap < 9; ++tap) {
    // ---- stage this tap's weight slice: w2[o][c][tap] -> Wlds[c*72 + o] ----
    if (tap) __syncthreads();                       // prior reads complete
    for (int e = tid; e < 3 * CH * CH; e += 256) {  // 12288 elements
      const int c = e >> 6;
      const int o = e & 63;
      async_g2l_b32(&Wlds[c * WSTRIDE + o], &w2[(o * (3 * CH) + c) * 9 + tap]);
    }
    wait_async0();
    __syncthreads();

    const int dx = tap / 3 - 1;
    const int dy = tap % 3 - 1;
    const int xx = x + dx;
    const int yy = y + dy;                          // per-lane column
    const bool valid = (xx >= 0) && (xx < WD) && (yy >= 0) && (yy < WD);
    const int p = valid ? (xx * WD + yy) : 0;

#pragma unroll 4
    for (int kc = 0; kc < 48; ++kc) {
      const int c0 = kc * 4 + half * 2;
      // Hoist all operand loads (global features + LDS weights) ahead of
      // the 4-WMMA group so DS/VMEM latencies overlap the XDL burst.
      v2f bv;
      bv.x = valid ? fused_ld(CvB, CkB, VB, c0,     p) : 0.0f;
      bv.y = valid ? fused_ld(CvB, CkB, VB, c0 + 1, p) : 0.0f;
      v2f av[4];
#pragma unroll
      for (int ot = 0; ot < 4; ++ot) {
        const int o = ot * 16 + lh;
        av[ot].x = Wlds[c0 * WSTRIDE + o];
        av[ot].y = Wlds[(c0 + 1) * WSTRIDE + o];
      }
#pragma unroll
      for (int ot = 0; ot < 4; ++ot)
        acc[ot] = wmma_f32_k4(av[ot], bv, acc[ot]);
    }
  }

#pragma unroll
  for (int ot = 0; ot < 4; ++ot) {
#pragma unroll
    for (int r = 0; r < 8; ++r) {
      const int o = ot * 16 + r + 8 * half;
      const float sc = gamma[o] * rsqrtf(var[o] + 1e-5f);
      float v = (acc[ot][r] + b2[o] - mean[o]) * sc + beta[o];
      Out[(b * CH + o) * HW + x * WD + y0 + lh] = fmaxf(v, 0.0f);
    }
  }
}

// ---------------------------------------------------------------------------
extern "C" void kernel_launch(void* const* d_in, const int* in_sizes, int n_in,
                              void* d_out, int out_size, void* d_ws, size_t ws_size,
                              hipStream_t stream)
{
  (void)in_sizes; (void)n_in; (void)out_size; (void)ws_size;

  const float* V  = (const float*)d_in[0];   // V_rgb
  const float* Kd = (const float*)d_in[1];   // K_dep
  const float* Q  = (const float*)d_in[2];   // Q_flo
  const float* w1 = (const float*)d_in[3];
  const float* b1 = (const float*)d_in[4];
  const float* w2 = (const float*)d_in[5];
  const float* b2 = (const float*)d_in[6];
  const float* gm = (const float*)d_in[7];
  const float* bt = (const float*)d_in[8];
  const float* mn = (const float*)d_in[9];
  const float* vr = (const float*)d_in[10];
  float* out = (float*)d_out;

  // Workspace layout (~8.25 MB total)
  char* ws = (char*)d_ws;
  float* Sq = (float*)ws;  ws += (size_t)BATCH * HW * sizeof(float);
  float* Sk = (float*)ws;  ws += (size_t)BATCH * HW * sizeof(float);
  int*   Iq = (int*)ws;    ws += (size_t)BATCH * HW * sizeof(int);
  int*   Ik = (int*)ws;    ws += (size_t)BATCH * HW * sizeof(int);
  float* Cv = (float*)ws;  ws += (size_t)BATCH * CH * HW * sizeof(float);
  float* Ck = (float*)ws;

  const dim3 blk(256);                 // 8 waves per block (wave32)
  const dim3 grd(BATCH * 256 / 8);     // 1024 wave-tiles / 8 waves

  hard_match_kernel<<<grd, blk, 0, stream>>>(Q,  V, Sq, Iq);
  hard_match_kernel<<<grd, blk, 0, stream>>>(Kd, V, Sk, Ik);
  conv1_fuse_kernel<<<grd, blk, 0, stream>>>(V, Q,  Iq, Sq, w1, b1, Cv);
  conv1_fuse_kernel<<<grd, blk, 0, stream>>>(V, Kd, Ik, Sk, w1, b1, Ck);
  conv2_bn_relu_kernel<<<grd, blk, 0, stream>>>(Cv, Ck, V, w2, b2, gm, bt, mn, vr, out);
}